// Attention_55241869361238
// MI455X (gfx1250) — compile-verified
//
#include <hip/hip_runtime.h>
#include <math.h>

// ---------------------------------------------------------------------------
// Fused causal attention with RoPE + ALiBi for MI455X (gfx1250, wave32).
//   B=2, T=2048, D=1024, H=16, Dh=64.
// Kernel 1: qkv = x@W + b, RoPE on q/k, scatter to Q/K/V [B,H,T,Dh] in d_ws.
//           W strip shared across the block via async-LDS double-buffering.
// Kernel 2: flash attention (online softmax) with V_WMMA_F32_16X16X4_F32.
// ---------------------------------------------------------------------------

typedef float v2f __attribute__((ext_vector_type(2)));
typedef float v8f __attribute__((ext_vector_type(8)));
typedef int   v4i __attribute__((vector_size(16)));   // matches async-LDS builtin

#define T_LEN 2048
#define D_MODEL 1024
#define N_HEADS 16
#define D_HEAD 64
#define TRI 3072              // 3*D_MODEL
#define KC 16                 // K-chunk staged in LDS per block
#define WPAD 68               // padded LDS row (272B: 16B-aligned, bank-skewed)

#if defined(__has_builtin)
#if __has_builtin(__builtin_amdgcn_global_load_async_to_lds_b128)
#define USE_ASYNC_LDS 1
#endif
#endif
#ifndef USE_ASYNC_LDS
#define USE_ASYNC_LDS 0
#endif

__device__ __forceinline__ v8f wmma4(v2f a, v2f b, v8f c) {
    // D = A(16x4 f32) x B(4x16 f32) + C(16x16 f32)
    return __builtin_amdgcn_wmma_f32_16x16x4_f32(
        /*neg_a=*/false, a, /*neg_b=*/false, b,
        /*c_mod=*/(short)0, c, /*reuse_a=*/false, /*reuse_b=*/false);
}

__device__ __forceinline__ void wave_lds_fence() {
    asm volatile("s_wait_dscnt 0x0" ::: "memory");
    __builtin_amdgcn_wave_barrier();
}

// ---------------------------------------------------------------------------
// Kernel 1: QKV projection + bias + RoPE.
// grid: (48 strips, BT/64), block: 128 (4 waves). All 4 waves share one
// 64-col W strip -> stage KC x 64 W chunks in LDS (async if available).
// ---------------------------------------------------------------------------
__global__ __launch_bounds__(128)
void qkv_rope_kernel(const float* __restrict__ x,
                     const float* __restrict__ W,
                     const float* __restrict__ bias,
                     float* __restrict__ Qb,
                     float* __restrict__ Kb,
                     float* __restrict__ Vb) {
    __shared__ float wt[2][KC][WPAD];

    const int tid  = threadIdx.x;
    const int lane = tid & 31;
    const int wv   = tid >> 5;
    const int n    = lane & 15;          // col-in-tile / row-in-A per lane
    const int hi   = lane >> 4;          // lane half selects K pair
    const int strip   = blockIdx.x;                  // 0..47
    const int rowTile = blockIdx.y * 4 + wv;         // 0..BT/16-1
    const int c0      = strip * 64;

    // Stage W[kc .. kc+KC) x [c0 .. c0+64) into wt[bb]: 256 float4, 2/thread.
    auto stage = [&](int bb, int kc) {
#pragma unroll
        for (int u = 0; u < 2; ++u) {
            const int f   = tid * 2 + u;          // 0..255
            const int row = f >> 4;
            const int cq  = (f & 15) * 4;
            const float* g = W + (size_t)(kc + row) * TRI + c0 + cq;
            float* l = &wt[bb][row][cq];
#if USE_ASYNC_LDS
            __builtin_amdgcn_global_load_async_to_lds_b128(
                (__attribute__((address_space(1))) v4i*)g,
                (__attribute__((address_space(3))) v4i*)l, 0, 0);
#else
            *(float4*)l = *(const float4*)g;
#endif
        }
    };

    v8f acc[4];
#pragma unroll
    for (int j = 0; j < 4; ++j)
#pragma unroll
        for (int e = 0; e < 8; ++e) acc[j][e] = 0.0f;

    const float* xrow = x + (size_t)(rowTile * 16 + n) * D_MODEL;

    stage(0, 0);
    for (int kc = 0; kc < D_MODEL; kc += KC) {
        const int cur = (kc >> 4) & 1;
#if USE_ASYNC_LDS
        asm volatile("s_wait_asynccnt 0x0" ::: "memory");
#endif
        __syncthreads();                          // chunk `cur` visible to all
        if (kc + KC < D_MODEL) stage(cur ^ 1, kc + KC);

#pragma unroll
        for (int kk = 0; kk < KC; kk += 4) {
            // A fragment: lane n holds row M=n; VGPR0=K(+2*hi), VGPR1=K(+2*hi+1)
            v2f a = *(const v2f*)(xrow + kc + kk + 2 * hi);
            const float* l0 = &wt[cur][kk + 2 * hi][n];
            const float* l1 = &wt[cur][kk + 2 * hi + 1][n];
#pragma unroll
            for (int j = 0; j < 4; ++j) {
                v2f bf;
                bf.x = l0[j * 16];
                bf.y = l1[j * 16];
                acc[j] = wmma4(a, bf, acc[j]);
            }
        }
        __syncthreads();                          // done reading wt[cur]
    }

    // ------------------------- epilogue -------------------------
    const int sidx = strip >> 4;     // 0=q 1=k 2=v
    const int h    = strip & 15;

#pragma unroll
    for (int j = 0; j < 4; ++j) {
        float bv = bias[sidx * D_MODEL + h * D_HEAD + j * 16 + n];
#pragma unroll
        for (int e = 0; e < 8; ++e) acc[j][e] += bv;
    }

    if (sidx < 2) {
        // RoPE: pair (d, d+32); acc[j] pairs with acc[j+2], d = j*16+n < 32.
#pragma unroll
        for (int j = 0; j < 2; ++j) {
            const float invf = powf(10000.0f, -(float)(j * 16 + n) * (1.0f / 32.0f));
#pragma unroll
            for (int e = 0; e < 8; ++e) {
                int r = rowTile * 16 + e + 8 * hi;   // global row in B*T
                int t = r & (T_LEN - 1);
                float sn, cs;
                sincosf((float)t * invf, &sn, &cs);
                float x1 = acc[j][e], x2 = acc[j + 2][e];
                acc[j][e]     = x1 * cs - x2 * sn;
                acc[j + 2][e] = x2 * cs + x1 * sn;
            }
        }
    }

    float* dst = (sidx == 0) ? Qb : (sidx == 1) ? Kb : Vb;
#pragma unroll
    for (int j = 0; j < 4; ++j)
#pragma unroll
        for (int e = 0; e < 8; ++e) {
            int r = rowTile * 16 + e + 8 * hi;
            int bb = r >> 11;                 // r / T_LEN
            int t  = r & (T_LEN - 1);
            dst[(((size_t)(bb * N_HEADS + h)) * T_LEN + t) * D_HEAD + j * 16 + n] = acc[j][e];
        }
}

// ---------------------------------------------------------------------------
// Kernel 2: flash attention. grid: (T/64, H, B), block 128 (4 waves).
// One wave per 16-row q tile of one (b,h). Causal + ALiBi, online softmax.
// ---------------------------------------------------------------------------
__global__ __launch_bounds__(128)
void flash_attn_kernel(const float* __restrict__ Qb,
                       const float* __restrict__ Kb,
                       const float* __restrict__ Vb,
                       float* __restrict__ out) {
    __shared__ float pb[4][16][17];      // per-wave P transpose staging

    const int lane = threadIdx.x & 31;
    const int wv   = threadIdx.x >> 5;
    const int n    = lane & 15;
    const int hi   = lane >> 4;

    const int rowTile = blockIdx.x * 4 + wv;   // t tile
    const int h = blockIdx.y;
    const int b = blockIdx.z;
    const int t0 = rowTile * 16;
    const size_t bh = (size_t)(b * N_HEADS + h) * T_LEN;

    const float* Qp = Qb + bh * D_HEAD;
    const float* Kp = Kb + bh * D_HEAD;
    const float* Vp = Vb + bh * D_HEAD;

    const float coeff = exp2f(-8.0f + (7.0f / 15.0f) * (float)h);

    // Preload Q tile as 16 A-fragments (K-dim = Dh = 64, 4 per fragment).
    v2f qa[16];
#pragma unroll
    for (int kk = 0; kk < 16; ++kk)
        qa[kk] = *(const v2f*)(Qp + (size_t)(t0 + n) * D_HEAD + kk * 4 + 2 * hi);

    v8f o[4];
#pragma unroll
    for (int j = 0; j < 4; ++j)
#pragma unroll
        for (int e = 0; e < 8; ++e) o[j][e] = 0.0f;
    float mrow[8], lrow[8];
#pragma unroll
    for (int e = 0; e < 8; ++e) { mrow[e] = -INFINITY; lrow[e] = 0.0f; }

    for (int jt = 0; jt <= rowTile; ++jt) {
        const int s0 = jt * 16;

        // Prefetch next s-tile of K and V while we crunch this one.
        if (jt < rowTile) {
            __builtin_prefetch(Kp + (size_t)(s0 + 16 + n) * D_HEAD + 2 * hi, 0, 0);
            __builtin_prefetch(Vp + (size_t)(s0 + 16 + n) * D_HEAD + 2 * hi, 0, 0);
        }

        // ---- S = Q K^T (16x16 tile), 16 x wmma over Dh ----
        v8f sacc;
#pragma unroll
        for (int e = 0; e < 8; ++e) sacc[e] = 0.0f;
#pragma unroll
        for (int kk = 0; kk < 16; ++kk) {
            // B = K^T: lane n holds column s=s0+n; contiguous d pair.
            v2f kb = *(const v2f*)(Kp + (size_t)(s0 + n) * D_HEAD + kk * 4 + 2 * hi);
            sacc = wmma4(qa[kk], kb, sacc);
        }

        // ---- mask + ALiBi + online softmax (rows live across 16-lane halves)
#pragma unroll
        for (int e = 0; e < 8; ++e) {
            const int t = t0 + e + 8 * hi;
            const int s = s0 + n;
            float logit = sacc[e];
            logit = (s <= t) ? (logit - coeff * (float)(t - s)) : -INFINITY;

            float rmax = logit;
#pragma unroll
            for (int off = 1; off < 16; off <<= 1)
                rmax = fmaxf(rmax, __shfl_xor(rmax, off, 32));
            const float mnew  = fmaxf(mrow[e], rmax);
            const float scale = expf(mrow[e] - mnew);
            const float p     = expf(logit - mnew);

            float rsum = p;
#pragma unroll
            for (int off = 1; off < 16; off <<= 1)
                rsum += __shfl_xor(rsum, off, 32);

            lrow[e] = lrow[e] * scale + rsum;
            mrow[e] = mnew;
            pb[wv][e + 8 * hi][n] = p;         // stage for D->A transpose
#pragma unroll
            for (int j = 0; j < 4; ++j) o[j][e] *= scale;
        }
        wave_lds_fence();

        // ---- O += P V : A-frags from LDS, B-frags from V ----
#pragma unroll
        for (int kk = 0; kk < 4; ++kk) {
            v2f pa;
            pa.x = pb[wv][n][kk * 4 + 2 * hi];
            pa.y = pb[wv][n][kk * 4 + 2 * hi + 1];
#pragma unroll
            for (int jd = 0; jd < 4; ++jd) {
                const float* vrow = Vp + (size_t)(s0 + kk * 4 + 2 * hi) * D_HEAD + jd * 16 + n;
                v2f vb;
                vb.x = vrow[0];
                vb.y = vrow[D_HEAD];           // next s row
                o[jd] = wmma4(pa, vb, o[jd]);
            }
        }
        wave_lds_fence();
    }

    // ---- normalize and store: out[b, t, h*64 + d] ----
#pragma unroll
    for (int e = 0; e < 8; ++e) {
        const float inv_l = 1.0f / lrow[e];
        const int t = t0 + e + 8 * hi;
#pragma unroll
        for (int jd = 0; jd < 4; ++jd) {
            out[((size_t)(b * T_LEN + t)) * D_MODEL + h * D_HEAD + jd * 16 + n] =
                o[jd][e] * inv_l;
        }
    }
}

// ---------------------------------------------------------------------------
extern "C" void kernel_launch(void* const* d_in, const int* in_sizes, int n_in,
                              void* d_out, int out_size, void* d_ws, size_t ws_size,
                              hipStream_t stream) {
    const float* x    = (const float*)d_in[0];   // (B,T,D) fp32
    // d_in[1] = mask (bool) — causal structure is applied analytically.
    const float* W    = (const float*)d_in[2];   // (D, 3D) fp32
    const float* bias = (const float*)d_in[3];   // (3D,)   fp32
    float* out = (float*)d_out;

    const int Bn = in_sizes[0] / (T_LEN * D_MODEL);      // = 2
    const size_t qkv_elems = (size_t)Bn * N_HEADS * T_LEN * D_HEAD;

    float* Qb = (float*)d_ws;
    float* Kb = Qb + qkv_elems;
    float* Vb = Kb + qkv_elems;

    // Kernel 1: (BT/16) row tiles x 48 col strips, 4 waves/block.
    dim3 g1(48, (unsigned)(Bn * T_LEN / 64), 1);
    qkv_rope_kernel<<<g1, dim3(128), 0, stream>>>(x, W, bias, Qb, Kb, Vb);

    // Kernel 2: T/16 row tiles (4 waves/block) x H x B.
    dim3 g2((unsigned)(T_LEN / 64), N_HEADS, (unsigned)Bn);
    flash_attn_kernel<<<g2, dim3(128), 0, stream>>>(Qb, Kb, Vb, out);
}